// GAT_1468878815266
// MI455X (gfx1250) — compile-verified
//
#include <hip/hip_runtime.h>
#include <math.h>

// ---------------- problem dims (fixed by reference) ----------------
#define BB 16    // batch
#define NN 512   // nodes
#define FF 128   // nfeat = nclass
#define HH 8     // heads
#define DD 32    // nhid
#define HD 256   // H*D
#define CC 128   // nclass

#define ALPHAF 0.2f
#define NEGV   -9.0e15f

typedef __attribute__((ext_vector_type(2))) float v2f;
typedef __attribute__((ext_vector_type(4))) float v4f;
typedef __attribute__((ext_vector_type(8))) float v8f;
typedef __attribute__((ext_vector_type(2))) int   v2i;
typedef __attribute__((ext_vector_type(4))) int   v4i;

// D(16x16,f32) = A(16x4,f32) x B(4x16,f32) + C  -- CDNA5 V_WMMA_F32_16X16X4_F32
static __device__ __forceinline__ v8f wmma4(v2f a, v2f b, v8f c) {
  return __builtin_amdgcn_wmma_f32_16x16x4_f32(false, a, false, b, (short)0, c,
                                               false, false);
}

// K-paired layouts: element (k, n) lives at [k/2][n][k&1] so the WMMA B
// operand {M[k][n], M[k+1][n]} is one aligned 8-byte load.
static __device__ __forceinline__ size_t pairIdx(int k, int n, int ncols) {
  return ((size_t)(k >> 1) * ncols + n) * 2 + (k & 1);
}

// =====================================================================
// Pack kernels: W1[H,F,D] -> W1p[H][F/2][D][2], W2[HD,C] -> W2p[HD/2][C][2]
// =====================================================================
__global__ void k_pack_w1(const float* __restrict__ W1, float* __restrict__ W1p)
{
  const int idx = blockIdx.x * blockDim.x + threadIdx.x;   // H*F*D
  if (idx >= HH * FF * DD) return;
  const int d  = idx % DD;
  const int k  = (idx / DD) % FF;
  const int hd = idx / (DD * FF);
  W1p[(size_t)hd * FF * DD + pairIdx(k, d, DD)] = W1[idx];
}

__global__ void k_pack_w2(const float* __restrict__ W2, float* __restrict__ W2p)
{
  const int idx = blockIdx.x * blockDim.x + threadIdx.x;   // HD*CC
  if (idx >= HD * CC) return;
  const int c = idx % CC;
  const int k = idx / CC;
  W2p[pairIdx(k, c, CC)] = W2[idx];
}

// =====================================================================
// Kernel 1: h[b,h] = x[b] (512x128) @ W1[h] (128x32); h stored K-paired
// over its row (j) index for the downstream att@h WMMA.
// =====================================================================
__global__ __launch_bounds__(128) void k_gemm_h(
    const float* __restrict__ x, const float* __restrict__ W1p,
    float* __restrict__ hp)
{
  const int wid  = threadIdx.x >> 5;
  const int lane = threadIdx.x & 31;
  const int wave = blockIdx.x * 4 + wid;   // B*H*(N/16) waves
  const int it   = wave & 31;
  const int bh   = wave >> 5;
  const int hd   = bh & 7;
  const int b    = bh >> 3;
  const int i0   = it << 4;
  const int m = lane & 15, half = lane >> 4;

  const float* arow = x   + ((size_t)b * NN + i0 + m) * FF;
  const float* wb   = W1p + (size_t)hd * FF * DD;

  v8f c0 = {}, c1 = {};
#pragma unroll 4
  for (int k0 = 0; k0 < FF; k0 += 4) {
    const int kb = k0 + 2 * half;                   // even
    v2f a  = *(const v2f*)(arow + kb);              // A[m][kb], A[m][kb+1]
    v2f b0 = *(const v2f*)(wb + ((size_t)(kb >> 1) * DD + m) * 2);
    v2f b1 = *(const v2f*)(wb + ((size_t)(kb >> 1) * DD + 16 + m) * 2);
    c0 = wmma4(a, b0, c0);
    c1 = wmma4(a, b1, c1);
  }
  // store K-paired over row index j = i0 + (8*half + v)
  float* ob = hp + (size_t)bh * NN * DD;
#pragma unroll
  for (int v = 0; v < 8; ++v) {
    const int j = i0 + half * 8 + v;
    ob[pairIdx(j, m, DD)]      = c0[v];
    ob[pairIdx(j, 16 + m, DD)] = c1[v];
  }
}

// =====================================================================
// Kernel 2: f1/f2 per (b,h,n) from K-paired h
// =====================================================================
__global__ void k_f12(const float* __restrict__ hp, const float* __restrict__ a1,
                      float* __restrict__ f1, float* __restrict__ f2)
{
  const int idx = blockIdx.x * blockDim.x + threadIdx.x;   // B*H*N rows
  if (idx >= BB * HH * NN) return;
  const int j  = idx % NN;
  const int bh = idx / NN;
  const int hd = bh & 7;
  const float* hr = hp + (size_t)bh * NN * DD
                       + ((size_t)(j >> 1) * DD) * 2 + (j & 1);
  const float* av = a1 + (size_t)hd * 2 * DD;
  float s1 = 0.f, s2 = 0.f;
#pragma unroll
  for (int d = 0; d < DD; ++d) {
    const float v = hr[2 * d];
    s1 += v * av[d];
    s2 += v * av[DD + d];
  }
  f1[idx] = s1;
  f2[idx] = s2;
}

// =====================================================================
// Kernel 3: fused masked softmax + att@h + ELU + head concat
// =====================================================================
__global__ __launch_bounds__(128) void k_att1(
    const float* __restrict__ hp, const int* __restrict__ adj,
    const float* __restrict__ f1, const float* __restrict__ f2,
    float* __restrict__ xc)
{
  __shared__ __align__(16) float s_f2[NN];
  const int wid  = threadIdx.x >> 5;
  const int lane = threadIdx.x & 31;
  const int wave = blockIdx.x * 4 + wid;
  const int it   = wave & 31;
  const int bh   = wave >> 5;           // same (b,h) for all 4 waves of block
  const int hd   = bh & 7;
  const int b    = bh >> 3;
  const int i0   = it << 4;
  const int m = lane & 15, half = lane >> 4;
  const int row = i0 + m;

  for (int j = threadIdx.x; j < NN; j += 128)
    s_f2[j] = f2[(size_t)bh * NN + j];
  __syncthreads();

  const int*  adjrow = adj + ((size_t)b * NN + row) * NN;
  const float f1m    = f1[(size_t)bh * NN + row];

  // passes 1-2: lane-half scans a contiguous 256-j range with 128-bit loads
  const int jlo = half * 256;
  float mx = -3.4e38f;
  for (int j = jlo; j < jlo + 256; j += 4) {
    v4i ad = *(const v4i*)(adjrow + j);
    v4f fv = *(const v4f*)(s_f2 + j);
#pragma unroll
    for (int q = 0; q < 4; ++q) {
      float s = f1m + fv[q];
      s = s > 0.f ? s : ALPHAF * s;
      s = ad[q] > 0 ? s : NEGV;
      mx = fmaxf(mx, s);
    }
  }
  mx = fmaxf(mx, __shfl_xor(mx, 16, 32));

  float sum = 0.f;
  for (int j = jlo; j < jlo + 256; j += 4) {
    v4i ad = *(const v4i*)(adjrow + j);
    v4f fv = *(const v4f*)(s_f2 + j);
#pragma unroll
    for (int q = 0; q < 4; ++q) {
      float s = f1m + fv[q];
      s = s > 0.f ? s : ALPHAF * s;
      s = ad[q] > 0 ? s : NEGV;
      sum += __expf(s - mx);
    }
  }
  sum += __shfl_xor(sum, 16, 32);
  const float inv = 1.0f / sum;

  // pass 3: att(16xN) @ h(NxD); A computed in-register, B via paired b64 loads
  const float* hb = hp + (size_t)bh * NN * DD;
  v8f c0 = {}, c1 = {};
  for (int j0 = 0; j0 < NN; j0 += 4) {
    const int ja = j0 + 2 * half;                  // even
    v2i ad = *(const v2i*)(adjrow + ja);
    v2f fp = *(const v2f*)(s_f2 + ja);
    float s0 = f1m + fp.x;
    s0 = s0 > 0.f ? s0 : ALPHAF * s0;
    s0 = ad.x > 0 ? s0 : NEGV;
    float s1 = f1m + fp.y;
    s1 = s1 > 0.f ? s1 : ALPHAF * s1;
    s1 = ad.y > 0 ? s1 : NEGV;
    v2f a;
    a.x = __expf(s0 - mx) * inv;
    a.y = __expf(s1 - mx) * inv;
    v2f b0 = *(const v2f*)(hb + ((size_t)(ja >> 1) * DD + m) * 2);
    v2f b1 = *(const v2f*)(hb + ((size_t)(ja >> 1) * DD + 16 + m) * 2);
    c0 = wmma4(a, b0, c0);
    c1 = wmma4(a, b1, c1);
  }

  // ELU + concat: xc[b][i][hd*D + d]  (row-major; A operand of next GEMM)
  float* ob = xc + ((size_t)b * NN + i0) * HD + hd * DD;
#pragma unroll
  for (int v = 0; v < 8; ++v) {
    const int r = half * 8 + v;
    float x0 = c0[v]; x0 = x0 > 0.f ? x0 : __expf(x0) - 1.f;
    float x1 = c1[v]; x1 = x1 > 0.f ? x1 : __expf(x1) - 1.f;
    ob[(size_t)r * HD + m]      = x0;
    ob[(size_t)r * HD + 16 + m] = x1;
  }
}

// =====================================================================
// Kernel 4: h2[b] = xc[b] (512x256) @ W2 (256x128); h2 stored K-paired
// =====================================================================
__global__ __launch_bounds__(128) void k_gemm_h2(
    const float* __restrict__ xc, const float* __restrict__ W2p,
    float* __restrict__ h2p)
{
  const int wid  = threadIdx.x >> 5;
  const int lane = threadIdx.x & 31;
  const int wave = blockIdx.x * 4 + wid;   // B*(N/16) waves
  const int b    = wave >> 5;
  const int i0   = (wave & 31) << 4;
  const int m = lane & 15, half = lane >> 4;

  const float* arow = xc + ((size_t)b * NN + i0 + m) * HD;
  v8f c[8] = {};
  for (int k0 = 0; k0 < HD; k0 += 4) {
    const int kb = k0 + 2 * half;                   // even
    v2f a = *(const v2f*)(arow + kb);
#pragma unroll
    for (int t = 0; t < 8; ++t) {
      v2f bb = *(const v2f*)(W2p + ((size_t)(kb >> 1) * CC + t * 16 + m) * 2);
      c[t] = wmma4(a, bb, c[t]);
    }
  }
  float* ob = h2p + (size_t)b * NN * CC;
#pragma unroll
  for (int t = 0; t < 8; ++t)
#pragma unroll
    for (int v = 0; v < 8; ++v) {
      const int j = i0 + half * 8 + v;
      ob[pairIdx(j, t * 16 + m, CC)] = c[t][v];
    }
}

// =====================================================================
// Kernel 5: g1/g2 per (b,n) from K-paired h2
// =====================================================================
__global__ void k_g12(const float* __restrict__ h2p, const float* __restrict__ a2,
                      float* __restrict__ g1, float* __restrict__ g2)
{
  const int idx = blockIdx.x * blockDim.x + threadIdx.x;   // B*N rows
  if (idx >= BB * NN) return;
  const int j = idx % NN;
  const int b = idx / NN;
  const float* hr = h2p + (size_t)b * NN * CC
                        + ((size_t)(j >> 1) * CC) * 2 + (j & 1);
  float s1 = 0.f, s2 = 0.f;
#pragma unroll 8
  for (int d = 0; d < CC; ++d) {
    const float v = hr[2 * d];
    s1 += v * a2[d];
    s2 += v * a2[CC + d];
  }
  g1[idx] = s1;
  g2[idx] = s2;
}

// =====================================================================
// Kernel 6: output attention: masked softmax + att2@h2 + ELU + residual
// =====================================================================
__global__ __launch_bounds__(128) void k_att2(
    const float* __restrict__ h2p, const int* __restrict__ adj,
    const float* __restrict__ g1, const float* __restrict__ g2,
    const float* __restrict__ x, float* __restrict__ out)
{
  __shared__ __align__(16) float s_g2[NN];
  const int wid  = threadIdx.x >> 5;
  const int lane = threadIdx.x & 31;
  const int wave = blockIdx.x * 4 + wid;   // B*(N/16)
  const int b    = wave >> 5;              // same b for all 4 waves of block
  const int i0   = (wave & 31) << 4;
  const int m = lane & 15, half = lane >> 4;
  const int row = i0 + m;

  for (int j = threadIdx.x; j < NN; j += 128)
    s_g2[j] = g2[(size_t)b * NN + j];
  __syncthreads();

  const int*  adjrow = adj + ((size_t)b * NN + row) * NN;
  const float g1m    = g1[(size_t)b * NN + row];

  const int jlo = half * 256;
  float mx = -3.4e38f;
  for (int j = jlo; j < jlo + 256; j += 4) {
    v4i ad = *(const v4i*)(adjrow + j);
    v4f fv = *(const v4f*)(s_g2 + j);
#pragma unroll
    for (int q = 0; q < 4; ++q) {
      float s = g1m + fv[q];
      s = s > 0.f ? s : ALPHAF * s;
      s = ad[q] > 0 ? s : NEGV;
      mx = fmaxf(mx, s);
    }
  }
  mx = fmaxf(mx, __shfl_xor(mx, 16, 32));

  float sum = 0.f;
  for (int j = jlo; j < jlo + 256; j += 4) {
    v4i ad = *(const v4i*)(adjrow + j);
    v4f fv = *(const v4f*)(s_g2 + j);
#pragma unroll
    for (int q = 0; q < 4; ++q) {
      float s = g1m + fv[q];
      s = s > 0.f ? s : ALPHAF * s;
      s = ad[q] > 0 ? s : NEGV;
      sum += __expf(s - mx);
    }
  }
  sum += __shfl_xor(sum, 16, 32);
  const float inv = 1.0f / sum;

  const float* hb = h2p + (size_t)b * NN * CC;
  v8f c[8] = {};
  for (int j0 = 0; j0 < NN; j0 += 4) {
    const int ja = j0 + 2 * half;                  // even
    v2i ad = *(const v2i*)(adjrow + ja);
    v2f fp = *(const v2f*)(s_g2 + ja);
    float s0 = g1m + fp.x;
    s0 = s0 > 0.f ? s0 : ALPHAF * s0;
    s0 = ad.x > 0 ? s0 : NEGV;
    float s1 = g1m + fp.y;
    s1 = s1 > 0.f ? s1 : ALPHAF * s1;
    s1 = ad.y > 0 ? s1 : NEGV;
    v2f a;
    a.x = __expf(s0 - mx) * inv;
    a.y = __expf(s1 - mx) * inv;
#pragma unroll
    for (int t = 0; t < 8; ++t) {
      v2f bb = *(const v2f*)(hb + ((size_t)(ja >> 1) * CC + t * 16 + m) * 2);
      c[t] = wmma4(a, bb, c[t]);
    }
  }

  const float* xr = x   + ((size_t)b * NN + i0) * FF;
  float*       ob = out + ((size_t)b * NN + i0) * FF;
#pragma unroll
  for (int t = 0; t < 8; ++t)
#pragma unroll
    for (int v = 0; v < 8; ++v) {
      const int r = half * 8 + v;
      const int col = t * 16 + m;
      float val = c[t][v];
      val = val > 0.f ? val : __expf(val) - 1.f;
      ob[(size_t)r * FF + col] = val + xr[(size_t)r * FF + col];
    }
}

// =====================================================================
extern "C" void kernel_launch(void* const* d_in, const int* in_sizes, int n_in,
                              void* d_out, int out_size, void* d_ws,
                              size_t ws_size, hipStream_t stream)
{
  (void)in_sizes; (void)n_in; (void)out_size; (void)ws_size;
  const float* x   = (const float*)d_in[0];   // [B,N,F]
  const int*   adj = (const int*)  d_in[1];   // [B,N,N]
  const float* W1  = (const float*)d_in[2];   // [H,F,D]
  const float* a1  = (const float*)d_in[3];   // [H,2D]
  const float* W2  = (const float*)d_in[4];   // [H*D,C]
  const float* a2  = (const float*)d_in[5];   // [2C]
  float* out = (float*)d_out;

  // workspace partition (~21.8 MB of floats)
  float* ws  = (float*)d_ws;
  float* W1p = ws;                                   // H*F*D
  float* W2p = W1p + (size_t)HH * FF * DD;           // HD*CC
  float* hp  = W2p + (size_t)HD * CC;                // B*H*N*D (K-paired)
  float* f1  = hp  + (size_t)BB * HH * NN * DD;      // B*H*N
  float* f2  = f1  + (size_t)BB * HH * NN;           // B*H*N
  float* xc  = f2  + (size_t)BB * HH * NN;           // B*N*HD
  float* h2p = xc  + (size_t)BB * NN * HD;           // B*N*C (K-paired)
  float* g1  = h2p + (size_t)BB * NN * CC;           // B*N
  float* g2  = g1  + (size_t)BB * NN;                // B*N

  k_pack_w1<<<(HH * FF * DD) / 256,      256, 0, stream>>>(W1, W1p);
  k_pack_w2<<<(HD * CC) / 256,           256, 0, stream>>>(W2, W2p);
  k_gemm_h <<<(BB * HH * (NN / 16)) / 4, 128, 0, stream>>>(x, W1p, hp);
  k_f12    <<<(BB * HH * NN) / 256,      256, 0, stream>>>(hp, a1, f1, f2);
  k_att1   <<<(BB * HH * (NN / 16)) / 4, 128, 0, stream>>>(hp, adj, f1, f2, xc);
  k_gemm_h2<<<(BB * (NN / 16)) / 4,      128, 0, stream>>>(xc, W2p, h2p);
  k_g12    <<<(BB * NN) / 256,           256, 0, stream>>>(h2p, a2, g1, g2);
  k_att2   <<<(BB * (NN / 16)) / 4,      128, 0, stream>>>(h2p, adj, g1, g2, x, out);
}